// DiscreteExactLoss_56693568307777
// MI455X (gfx1250) — compile-verified
//
#include <hip/hip_runtime.h>
#include <cmath>

typedef __attribute__((ext_vector_type(2))) float v2f;
typedef __attribute__((ext_vector_type(8))) float v8f;

#define RECEPTORS 16
#define GROUPS    4
#define STATES    16   // 2^4 joint binary states per 4-receptor group
#define VCOLS     64   // GROUPS * STATES

// ---------------------------------------------------------------------------
// Stage 0: zero the fp64 accumulator (graph replays don't re-poison d_ws,
// so it must be re-initialized every call).
// ---------------------------------------------------------------------------
__global__ void dex_zero(double* __restrict__ acc) { *acc = 0.0; }

// ---------------------------------------------------------------------------
// Stage 1: expand activity[B,16] -> V[B,64]: for each group g of 4 receptors,
// the 16 joint-state probabilities  prod_j (bit_j(s) ? a_j : 1-a_j).
// Then  P[b,c] = prod_r match_r[b,c]  ==  prod_g (V_g V_g^T)[b,c].
// ---------------------------------------------------------------------------
__global__ void dex_build_states(const float* __restrict__ act,
                                 float* __restrict__ V, int B) {
  int row = blockIdx.x * blockDim.x + threadIdx.x;
  if (row >= B) return;
  float a[RECEPTORS];
#pragma unroll
  for (int r = 0; r < RECEPTORS; ++r) a[r] = act[(size_t)row * RECEPTORS + r];
  float* vrow = V + (size_t)row * VCOLS;
#pragma unroll
  for (int g = 0; g < GROUPS; ++g) {
    float p[STATES];
    p[0] = 1.0f;
#pragma unroll
    for (int j = 0; j < 4; ++j) {
      float aj = a[g * 4 + j];
      float bj = 1.0f - aj;
#pragma unroll
      for (int s = 0; s < (1 << 3); ++s) {
        if (s < (1 << j)) {
          float base = p[s];
          p[s + (1 << j)] = base * aj;
          p[s]            = base * bj;
        }
      }
    }
#pragma unroll
    for (int s = 0; s < STATES; ++s) vrow[g * STATES + s] = p[s];
  }
}

// ---------------------------------------------------------------------------
// Stage 2: one wave per 16x16 pair-tile, upper triangle only (P symmetric:
// off-diagonal tiles weighted 2x; diagonal tiles masked at b==c, weight 1x).
//
// For each group g: Gram tile P_g = A B^T via 4 chained V_WMMA_F32_16X16X4_F32
// (K=16). All 32 float2 fragments are loaded up-front into distinct registers
// so the loads batch into one clause and the 16 WMMAs issue without
// interleaved s_wait_loadcnt stalls. Then elementwise product across groups,
// diagonal mask, fp64 wave butterfly, one global fp64 atomic per tile.
//
// f32 WMMA 16x16x4 fragment layout (wave32):
//   A: lane l holds A[l%16, v + 2*(l>=16)]   (v = VGPR index 0..1)
//   B: lane l holds B[v + 2*(l>=16), l%16]
// Symmetric Gram => both sides load with the identical per-lane pattern:
// aligned float2 at  row*64 + g*16 + 4*chunk + 2*(lane>=16).
//   D: element i of the v8f accumulator is (M = i + 8*(lane>=16), N = l%16).
// ---------------------------------------------------------------------------
__global__ void __launch_bounds__(32)
dex_tile_sum(const float* __restrict__ V, double* __restrict__ acc) {
  const int bt = blockIdx.x;
  const int ct = blockIdx.y;
  if (bt > ct) return;                 // uniform whole-wave exit (EXEC intact)

  const int lane = threadIdx.x;        // 0..31, full wave active
  const int half = lane >> 4;          // 0 or 1
  const int lrow = lane & 15;

  const float* __restrict__ arow = V + (size_t)(bt * 16 + lrow) * VCOLS;
  const float* __restrict__ brow = V + (size_t)(ct * 16 + lrow) * VCOLS;

  // Hoist all fragment loads (4 groups x 4 k-chunks x 2 sides = 32 x b64).
  v2f af[GROUPS][4], bf[GROUPS][4];
#pragma unroll
  for (int g = 0; g < GROUPS; ++g) {
#pragma unroll
    for (int ch = 0; ch < 4; ++ch) {
      const int koff = g * STATES + ch * 4 + 2 * half;
      af[g][ch] = *(const v2f*)(arow + koff);
      bf[g][ch] = *(const v2f*)(brow + koff);
    }
  }

  v8f pg[GROUPS];
#pragma unroll
  for (int g = 0; g < GROUPS; ++g) {
    v8f c = {};
#pragma unroll
    for (int ch = 0; ch < 4; ++ch) {
      // 8 args: (neg_a, A, neg_b, B, c_mod, C, reuse_a, reuse_b)
      c = __builtin_amdgcn_wmma_f32_16x16x4_f32(
          false, af[g][ch], false, bf[g][ch], (short)0, c, false, false);
    }
    pg[g] = c;
  }

  // P = P0 * P1 * P2 * P3 elementwise; mask self-pairs on diagonal tiles.
  const bool  diag = (bt == ct);
  const double w   = diag ? 1.0 : 2.0;   // symmetry weight
  double s = 0.0;
#pragma unroll
  for (int i = 0; i < 8; ++i) {
    float p = pg[0][i] * pg[1][i] * pg[2][i] * pg[3][i];
    if (diag && (i + 8 * half) == lrow) p = 0.0f;
    s += (double)p;
  }
  s *= w;

  // wave32 butterfly reduction in fp64
#pragma unroll
  for (int off = 16; off > 0; off >>= 1)
    s += __shfl_xor(s, off, 32);

  if (lane == 0)
    unsafeAtomicAdd(acc, s);   // global_atomic_add_f64
}

// ---------------------------------------------------------------------------
// Stage 3: answer = (ln(sum_offdiag P) - ln(B*(B-1))) / ln 2
// ---------------------------------------------------------------------------
__global__ void dex_finalize(const double* __restrict__ acc,
                             float* __restrict__ out, double log_num_pairs) {
  const double INV_LN2 = 1.4426950408889634;
  out[0] = (float)((log(*acc) - log_num_pairs) * INV_LN2);
}

extern "C" void kernel_launch(void* const* d_in, const int* in_sizes, int n_in,
                              void* d_out, int out_size, void* d_ws, size_t ws_size,
                              hipStream_t stream) {
  const float* act = (const float*)d_in[0];
  const int B = in_sizes[0] / RECEPTORS;          // 2048
  double* acc = (double*)d_ws;                    // 8B accumulator
  float*  V   = (float*)((char*)d_ws + 256);      // state table, B*64 floats
  float*  out = (float*)d_out;

  dex_zero<<<1, 1, 0, stream>>>(acc);
  dex_build_states<<<(B + 255) / 256, 256, 0, stream>>>(act, V, B);

  const int T = B / 16;                           // 128 tiles per dim
  dex_tile_sum<<<dim3(T, T), 32, 0, stream>>>(V, acc);

  const double log_np = std::log((double)B * (double)(B - 1));
  dex_finalize<<<1, 1, 0, stream>>>(acc, out, log_np);
}